// _SparseMoEClassifierLayer_56556129354192
// MI455X (gfx1250) — compile-verified
//
#include <hip/hip_runtime.h>
#include <math.h>

#define D_MODEL 1024
#define HIDDEN  4096
#define N_EXPERTS 8
#define NTOK    8192
#define LN_EPS  1e-5f

#define MROWS 32                           // rows per block (2 WMMA M-tiles)
#define KQ    256                          // X K-quarter staged in LDS
#define XS    268                          // LDS stride for X (256+12): conflict-free, 16B-aligned rows
#define HS    132                          // LDS stride for H chunk (128+4): conflict-free
#define ROWS_MAX (NTOK*2 + N_EXPERTS*MROWS)   // padded segment upper bound

typedef __attribute__((ext_vector_type(2))) float v2f;
typedef __attribute__((ext_vector_type(8))) float v8f;
typedef __attribute__((ext_vector_type(4))) float v4f;

// ---------------------------------------------------------------- init
__global__ void moe_init(float* mixed, int* counts, int* cursors, int* rowtok) {
    size_t i = (size_t)blockIdx.x * blockDim.x + threadIdx.x;
    v4f z = {0.f, 0.f, 0.f, 0.f};
    if (i * 4 < (size_t)NTOK * D_MODEL) ((v4f*)mixed)[i] = z;
    if (blockIdx.x == 0) {
        if (threadIdx.x < N_EXPERTS) { counts[threadIdx.x] = 0; cursors[threadIdx.x] = 0; }
        for (int r = threadIdx.x; r < ROWS_MAX; r += 256) rowtok[r] = -1;
    }
}

// ---------------------------------------------------------------- router (one wave per token)
__global__ void moe_router(const float* __restrict__ x, const float* __restrict__ gw,
                           float* __restrict__ probs, float* __restrict__ gates,
                           int* __restrict__ topidx, int* __restrict__ counts) {
    const int lane = threadIdx.x & 31;
    const int wv   = threadIdx.x >> 5;
    const int t    = blockIdx.x * 8 + wv;
    const float* xr = x + (size_t)t * D_MODEL;
    float acc[N_EXPERTS];
#pragma unroll
    for (int e = 0; e < N_EXPERTS; ++e) acc[e] = 0.f;
    for (int d = lane; d < D_MODEL; d += 32) {
        float xv = xr[d];
        const float* gr = gw + (size_t)d * N_EXPERTS;
#pragma unroll
        for (int e = 0; e < N_EXPERTS; ++e) acc[e] += xv * gr[e];
    }
#pragma unroll
    for (int e = 0; e < N_EXPERTS; ++e)
        for (int off = 16; off > 0; off >>= 1)
            acc[e] += __shfl_down(acc[e], off, 32);
    if (lane == 0) {
        float m = acc[0];
#pragma unroll
        for (int e = 1; e < N_EXPERTS; ++e) m = fmaxf(m, acc[e]);
        float p[N_EXPERTS], s = 0.f;
#pragma unroll
        for (int e = 0; e < N_EXPERTS; ++e) { p[e] = __expf(acc[e] - m); s += p[e]; }
        float inv = 1.f / s;
#pragma unroll
        for (int e = 0; e < N_EXPERTS; ++e) { p[e] *= inv; probs[(size_t)t*N_EXPERTS + e] = p[e]; }
        int i0 = 0; float b0 = p[0];
#pragma unroll
        for (int e = 1; e < N_EXPERTS; ++e) if (p[e] > b0) { b0 = p[e]; i0 = e; }
        int i1 = -1; float b1v = -1.f;
#pragma unroll
        for (int e = 0; e < N_EXPERTS; ++e) if (e != i0 && p[e] > b1v) { b1v = p[e]; i1 = e; }
        float sg = b0 + b1v;
        gates[t*2 + 0] = b0 / sg;  gates[t*2 + 1] = b1v / sg;
        topidx[t*2 + 0] = i0;      topidx[t*2 + 1] = i1;
        atomicAdd(&counts[i0], 1); atomicAdd(&counts[i1], 1);
    }
}

// ---------------------------------------------------------------- segments (pad to MROWS)
__global__ void moe_segments(const int* __restrict__ counts, int* __restrict__ seg) {
    if (threadIdx.x == 0 && blockIdx.x == 0) {
        int acc = 0;
#pragma unroll
        for (int e = 0; e < N_EXPERTS; ++e) { seg[e] = acc; acc += (counts[e] + MROWS-1) & ~(MROWS-1); }
        seg[N_EXPERTS] = acc;
    }
}

// ---------------------------------------------------------------- assign
__global__ void moe_assign(const int* __restrict__ topidx, const float* __restrict__ gates,
                           const int* __restrict__ seg, int* __restrict__ cursors,
                           int* __restrict__ rowtok, float* __restrict__ rowgate) {
    int t = blockIdx.x * 256 + threadIdx.x;
    if (t >= NTOK) return;
#pragma unroll
    for (int k = 0; k < 2; ++k) {
        int e   = topidx[t*2 + k];
        int pos = atomicAdd(&cursors[e], 1);
        int row = seg[e] + pos;
        rowtok[row]  = t;
        rowgate[row] = gates[t*2 + k];
    }
}

// ---------------------------------------------------------------- fused expert MLP (WMMA fp32)
// block: 256 threads (8 waves), 32 rows x 1024 out cols; hidden in chunks of 128.
// B fragments double-buffered in registers: 16 loads in flight per 16 WMMAs.
__global__ __launch_bounds__(256) void moe_mlp(
        const float* __restrict__ x,  const float* __restrict__ w1, const float* __restrict__ b1,
        const float* __restrict__ w2, const float* __restrict__ b2,
        const int* __restrict__ seg,  const int* __restrict__ rowtok,
        const float* __restrict__ rowgate, float* __restrict__ mixed) {
    __shared__ float lds_x[MROWS * XS];     // 34304 B
    __shared__ float lds_h[MROWS * HS];     // 16896 B
    __shared__ int   toks_s[MROWS];
    __shared__ float gts_s[MROWS];
    __shared__ int   einfo[2];

    const int row_base = blockIdx.x * MROWS;
    if (threadIdx.x == 0) {
        int total = seg[N_EXPERTS];
        int e = -1;
        if (row_base < total) {
#pragma unroll
            for (int q = 0; q < N_EXPERTS; ++q)
                if (row_base >= seg[q] && row_base < seg[q + 1]) e = q;
        }
        einfo[0] = e; einfo[1] = (e >= 0);
    }
    if (threadIdx.x < MROWS) {
        int row = row_base + threadIdx.x;
        int tk  = (row < ROWS_MAX) ? rowtok[row] : -1;
        toks_s[threadIdx.x] = tk;
        gts_s[threadIdx.x]  = (tk >= 0) ? rowgate[row] : 0.f;
    }
    __syncthreads();
    if (!einfo[1]) return;
    const int expert = einfo[0];

    const float* w1e = w1 + (size_t)expert * D_MODEL * HIDDEN;
    const float* b1e = b1 + (size_t)expert * HIDDEN;
    const float* w2e = w2 + (size_t)expert * HIDDEN * D_MODEL;
    const float* b2e = b2 + (size_t)expert * D_MODEL;

    const int lane = threadIdx.x & 31;
    const int wv   = threadIdx.x >> 5;
    const int half = lane >> 4;
    const int ll   = lane & 15;

    v8f cacc[16];                           // 2 M-tiles x 8 N-tiles
#pragma unroll
    for (int t = 0; t < 16; ++t) cacc[t] = (v8f){0.f,0.f,0.f,0.f,0.f,0.f,0.f,0.f};

    v8f hacc0, hacc1;

    // ---- stage-1 helpers: batch-load 8 B pairs / run 16 WMMAs on a k-group of 32
    const int hcol_off = wv * 16 + ll;
    auto ldB1 = [&](v2f* bb, int kbase, int ncol) {
#pragma unroll
        for (int j = 0; j < 8; ++j) {
            int gk = kbase + 4 * j + 2 * half;
            bb[j].x = w1e[(size_t)gk       * HIDDEN + ncol];
            bb[j].y = w1e[(size_t)(gk + 1) * HIDDEN + ncol];
        }
    };
    auto mma1 = [&](const v2f* bb, int kg) {
#pragma unroll
        for (int j = 0; j < 8; ++j) {
            int k = kg + 4 * j;
            v2f a0, a1;
            a0.x = lds_x[ll * XS + k + 2 * half];
            a0.y = lds_x[ll * XS + k + 1 + 2 * half];
            a1.x = lds_x[(16 + ll) * XS + k + 2 * half];
            a1.y = lds_x[(16 + ll) * XS + k + 1 + 2 * half];
            hacc0 = __builtin_amdgcn_wmma_f32_16x16x4_f32(false, a0, false, bb[j],
                                                          (short)0, hacc0, false, false);
            hacc1 = __builtin_amdgcn_wmma_f32_16x16x4_f32(false, a1, false, bb[j],
                                                          (short)0, hacc1, false, false);
        }
    };
    // ---- stage-2 helpers
    auto ldB2 = [&](v2f* bb, int gkbase) {
        const float* r0 = w2e + (size_t)(gkbase + 2 * half) * D_MODEL + wv * 128 + ll;
        const float* r1 = r0 + D_MODEL;
#pragma unroll
        for (int j = 0; j < 8; ++j) { bb[j].x = r0[j * 16]; bb[j].y = r1[j * 16]; }
    };
    auto mma2 = [&](const v2f* bb, int kk) {
        v2f a0, a1;
        a0.x = lds_h[ll * HS + kk + 2 * half];
        a0.y = lds_h[ll * HS + kk + 1 + 2 * half];
        a1.x = lds_h[(16 + ll) * HS + kk + 2 * half];
        a1.y = lds_h[(16 + ll) * HS + kk + 1 + 2 * half];
#pragma unroll
        for (int j = 0; j < 8; ++j) {
            cacc[j]     = __builtin_amdgcn_wmma_f32_16x16x4_f32(false, a0, false, bb[j],
                                                                (short)0, cacc[j], false, false);
            cacc[8 + j] = __builtin_amdgcn_wmma_f32_16x16x4_f32(false, a1, false, bb[j],
                                                                (short)0, cacc[8 + j], false, false);
        }
    };

    v2f bA[8], bB[8];

    for (int hc = 0; hc < HIDDEN; hc += 128) {
        // -------- stage 1: H[32 x 128] = gelu(X @ W1[:, hc:hc+128] + b1)
        hacc0 = (v8f){0.f,0.f,0.f,0.f,0.f,0.f,0.f,0.f};
        hacc1 = (v8f){0.f,0.f,0.f,0.f,0.f,0.f,0.f,0.f};
        const int ncol = hc + hcol_off;
        for (int kb = 0; kb < D_MODEL; kb += KQ) {
            __syncthreads();
            ldB1(bA, kb, ncol);                     // overlap with X staging
            // stage X quarter: 32 rows x 256 cols, vectorized b128 fills
            for (int i = threadIdx.x; i < MROWS * (KQ/4); i += 256) {
                int m = i >> 6, kq = (i & 63) << 2;
                int tk = toks_s[m]; if (tk < 0) tk = 0;
                v4f xv = *(const v4f*)&x[(size_t)tk * D_MODEL + kb + kq];
                *(v4f*)&lds_x[m * XS + kq] = xv;
            }
            __syncthreads();
#pragma unroll
            for (int kg = 0; kg < KQ; kg += 64) {
                ldB1(bB, kb + kg + 32, ncol);
                mma1(bA, kg);
                int nk = (kg + 64 < KQ) ? (kb + kg + 64) : kb;   // clamped (dummy on tail)
                ldB1(bA, nk, ncol);
                mma1(bB, kg + 32);
            }
        }
        {   // bias + gelu(tanh approx) -> LDS  (safe: all waves passed this chunk's barriers)
            float bias = b1e[hc + hcol_off];
#pragma unroll
            for (int v = 0; v < 8; ++v) {
                float hv = hacc0[v] + bias;
                float u  = 0.7978845608028654f * (hv + 0.044715f * hv * hv * hv);
                lds_h[(v + 8 * half) * HS + hcol_off] = 0.5f * hv * (1.0f + tanhf(u));
                float hw = hacc1[v] + bias;
                float u2 = 0.7978845608028654f * (hw + 0.044715f * hw * hw * hw);
                lds_h[(16 + v + 8 * half) * HS + hcol_off] = 0.5f * hw * (1.0f + tanhf(u2));
            }
        }
        ldB2(bA, hc);                               // prefetch overlaps the barrier
        __syncthreads();
        // -------- stage 2: OUT[32 x 1024] += H @ W2[hc:hc+128, :]
        for (int kk = 0; kk < 128; kk += 8) {
            ldB2(bB, hc + kk + 4);
            mma2(bA, kk);
            int nk = (kk + 8 < 128) ? (hc + kk + 8) : hc;        // clamped (dummy on tail)
            ldB2(bA, nk);
            mma2(bB, kk + 4);
        }
        // lds_h only overwritten after the barriers at the top of the next chunk
    }

    // epilogue: gate-weighted scatter-add (exactly 2 commutative adds per output -> deterministic)
#pragma unroll
    for (int t = 0; t < 8; ++t) {
        int col  = wv * 128 + t * 16 + ll;
        float bb = b2e[col];
#pragma unroll
        for (int v = 0; v < 8; ++v) {
            int m0 = v + 8 * half;
            int m1 = 16 + m0;
            int tk0 = toks_s[m0], tk1 = toks_s[m1];
            if (tk0 >= 0)
                unsafeAtomicAdd(&mixed[(size_t)tk0 * D_MODEL + col], gts_s[m0] * (cacc[t][v] + bb));
            if (tk1 >= 0)
                unsafeAtomicAdd(&mixed[(size_t)tk1 * D_MODEL + col], gts_s[m1] * (cacc[8 + t][v] + bb));
        }
    }
}

// ---------------------------------------------------------------- combine + layernorm
__global__ void moe_combine_ln(const float* __restrict__ x, const float* __restrict__ mixed,
                               const float* __restrict__ ln_g, const float* __restrict__ ln_b,
                               float* __restrict__ y) {
    __shared__ float red[256];
    const int t   = blockIdx.x;
    const int tid = threadIdx.x;
    float h[4]; float s = 0.f;
#pragma unroll
    for (int i = 0; i < 4; ++i) {
        int d = tid + i * 256;
        h[i]  = x[(size_t)t * D_MODEL + d] + mixed[(size_t)t * D_MODEL + d];
        s += h[i];
    }
    red[tid] = s; __syncthreads();
    for (int off = 128; off > 0; off >>= 1) { if (tid < off) red[tid] += red[tid + off]; __syncthreads(); }
    float mu = red[0] * (1.f / D_MODEL);
    __syncthreads();
    float vs = 0.f;
#pragma unroll
    for (int i = 0; i < 4; ++i) { float d = h[i] - mu; vs += d * d; }
    red[tid] = vs; __syncthreads();
    for (int off = 128; off > 0; off >>= 1) { if (tid < off) red[tid] += red[tid + off]; __syncthreads(); }
    float inv = rsqrtf(red[0] * (1.f / D_MODEL) + LN_EPS);
#pragma unroll
    for (int i = 0; i < 4; ++i) {
        int d = tid + i * 256;
        y[(size_t)t * D_MODEL + d] = (h[i] - mu) * inv * ln_g[d] + ln_b[d];
    }
}

// ---------------------------------------------------------------- aux loss (deterministic)
__global__ void moe_aux(const float* __restrict__ probs, const int* __restrict__ counts,
                        float* __restrict__ out_aux) {
    __shared__ float red[256];
    __shared__ float auxs;
    const int tid = threadIdx.x;
    if (tid == 0) auxs = 0.f;
    __syncthreads();
    for (int e = 0; e < N_EXPERTS; ++e) {
        float s = 0.f;
        for (int t = tid; t < NTOK; t += 256) s += probs[(size_t)t * N_EXPERTS + e];
        red[tid] = s; __syncthreads();
        for (int off = 128; off > 0; off >>= 1) { if (tid < off) red[tid] += red[tid + off]; __syncthreads(); }
        if (tid == 0) auxs += (red[0] / (float)NTOK) * ((float)counts[e] / (float)NTOK);
        __syncthreads();
    }
    if (tid == 0) *out_aux = (float)N_EXPERTS * auxs;
}

// ---------------------------------------------------------------- launch
extern "C" void kernel_launch(void* const* d_in, const int* in_sizes, int n_in,
                              void* d_out, int out_size, void* d_ws, size_t ws_size,
                              hipStream_t stream) {
    const float* x    = (const float*)d_in[0];
    const float* gw   = (const float*)d_in[1];
    const float* w1   = (const float*)d_in[2];
    const float* b1   = (const float*)d_in[3];
    const float* w2   = (const float*)d_in[4];
    const float* b2   = (const float*)d_in[5];
    const float* ln_g = (const float*)d_in[6];
    const float* ln_b = (const float*)d_in[7];
    float* out = (float*)d_out;

    char* p = (char*)d_ws;
    auto take = [&](size_t bytes) { char* r = p; p += (bytes + 255) & ~(size_t)255; return r; };
    float* mixed   = (float*)take((size_t)NTOK * D_MODEL * sizeof(float));
    float* probs   = (float*)take((size_t)NTOK * N_EXPERTS * sizeof(float));
    float* gates   = (float*)take((size_t)NTOK * 2 * sizeof(float));
    int*   topidx  = (int*)  take((size_t)NTOK * 2 * sizeof(int));
    int*   counts  = (int*)  take(N_EXPERTS * sizeof(int));
    int*   cursors = (int*)  take(N_EXPERTS * sizeof(int));
    int*   seg     = (int*)  take((N_EXPERTS + 1) * sizeof(int));
    int*   rowtok  = (int*)  take(ROWS_MAX * sizeof(int));
    float* rowgate = (float*)take(ROWS_MAX * sizeof(float));

    moe_init<<<(NTOK * D_MODEL) / (256 * 4), 256, 0, stream>>>(mixed, counts, cursors, rowtok);
    moe_router<<<NTOK / 8, 256, 0, stream>>>(x, gw, probs, gates, topidx, counts);
    moe_segments<<<1, 32, 0, stream>>>(counts, seg);
    moe_assign<<<NTOK / 256, 256, 0, stream>>>(topidx, gates, seg, cursors, rowtok, rowgate);
    moe_mlp<<<ROWS_MAX / MROWS, 256, 0, stream>>>(x, w1, b1, w2, b2, seg, rowtok, rowgate, mixed);
    moe_combine_ln<<<NTOK, 256, 0, stream>>>(x, mixed, ln_g, ln_b, out);
    moe_aux<<<1, 256, 0, stream>>>(probs, counts, out + (size_t)NTOK * D_MODEL);
}